// GATPlanner_70377334112974
// MI455X (gfx1250) — compile-verified
//
#include <hip/hip_runtime.h>
#include <math.h>

// ---------------------------------------------------------------------------
// GATPlanner forward for MI455X (gfx1250, wave32).
// All GEMMs run on V_WMMA_F32_16X16X4_F32 (full fp32 precision through the
// 6-layer softmax chain). Attention is fused flash-style: 16 score rows live
// in LDS between the QK^T WMMA, the masked softmax, the aij*X propagation
// WMMA and the wf feature-mix WMMA, so the 32 x (512x512) attention matrices
// never touch memory. Working set ~40MB -> resident in 192MB L2.
// All strides are template constants so per-lane stride-1 load pairs merge
// into b64 loads; waves carry 4 independent WMMA chains for XDL pipelining.
// ---------------------------------------------------------------------------

#define PP     4         // P heads
#define NN     512       // graph nodes
#define LSLOPE 0.2f
#define SST    516       // scores LDS row stride (516%64==4 -> conflict-free)
#define SZT    20        // z LDS row stride (halves land 32 banks apart)
#define GCH    128       // g-chunk staged in LDS per iteration

typedef __attribute__((ext_vector_type(2))) float v2f;
typedef __attribute__((ext_vector_type(8))) float v8f;

__device__ __forceinline__ v8f v8zero() {
  v8f z = {0.f, 0.f, 0.f, 0.f, 0.f, 0.f, 0.f, 0.f};
  return z;
}

__device__ __forceinline__ v8f wmma4(v2f a, v2f b, v8f c) {
  return __builtin_amdgcn_wmma_f32_16x16x4_f32(
      false, a, false, b, (short)0, c, false, false);
}

// ISA f32 16x16x4 fragment layout (wave32): lane idx = M (A) / N (B) index,
// half-wave selects K pair; VGPR j holds K = 2*half + j.
// A[m][k] at A[m*as_m + k*as_k]; B[k][n] at B[k*bs_k + n*bs_n].

// --- single tile, one accumulator chain ------------------------------------
__device__ __forceinline__ v8f wmma_tile(
    const float* __restrict__ A, long as_m, long as_k,
    const float* __restrict__ B, long bs_k, long bs_n,
    int Ktot, v8f acc)
{
  const int lane = threadIdx.x & 31;
  const int idx  = lane & 15;
  const int half = lane >> 4;
  for (int k = 0; k < Ktot; k += 4) {
    const int ka = k + 2 * half;
    v2f a, b;
    a.x = A[(long)idx * as_m + (long)(ka + 0) * as_k];
    a.y = A[(long)idx * as_m + (long)(ka + 1) * as_k];
    b.x = B[(long)(ka + 0) * bs_k + (long)idx * bs_n];
    b.y = B[(long)(ka + 1) * bs_k + (long)idx * bs_n];
    acc = wmma4(a, b, acc);
  }
  return acc;
}

// --- single tile, K split in two interleaved chains (long-K GEMMs) ---------
__device__ __forceinline__ v8f wmma_tile_split2(
    const float* __restrict__ A, long as_m, long as_k,
    const float* __restrict__ B, long bs_k, long bs_n,
    int Ktot, v8f acc)
{
  const int lane = threadIdx.x & 31;
  const int idx  = lane & 15;
  const int half = lane >> 4;
  const int h = Ktot >> 1;
  v8f acc1 = v8zero();
  for (int k = 0; k < h; k += 4) {
    const int ka = k + 2 * half;
    v2f a, b;
    a.x = A[(long)idx * as_m + (long)(ka + 0) * as_k];
    a.y = A[(long)idx * as_m + (long)(ka + 1) * as_k];
    b.x = B[(long)(ka + 0) * bs_k + (long)idx * bs_n];
    b.y = B[(long)(ka + 1) * bs_k + (long)idx * bs_n];
    acc = wmma4(a, b, acc);
    const int kb = ka + h;
    a.x = A[(long)idx * as_m + (long)(kb + 0) * as_k];
    a.y = A[(long)idx * as_m + (long)(kb + 1) * as_k];
    b.x = B[(long)(kb + 0) * bs_k + (long)idx * bs_n];
    b.y = B[(long)(kb + 1) * bs_k + (long)idx * bs_n];
    acc1 = wmma4(a, b, acc1);
  }
  return acc + acc1;
}

// --- one A fragment feeding 4 B tiles: 4 independent chains, A reuse 4x ----
__device__ __forceinline__ void wmma_tile4(
    const float* __restrict__ A, long as_m, long as_k,
    const float* __restrict__ B, long bs_k, long bs_n, long btile,
    int Ktot, v8f acc[4])
{
  const int lane = threadIdx.x & 31;
  const int idx  = lane & 15;
  const int half = lane >> 4;
  for (int k = 0; k < Ktot; k += 4) {
    const int ka = k + 2 * half;
    v2f a;
    a.x = A[(long)idx * as_m + (long)(ka + 0) * as_k];
    a.y = A[(long)idx * as_m + (long)(ka + 1) * as_k];
#pragma unroll
    for (int j = 0; j < 4; ++j) {
      const float* Bj = B + (long)j * btile;
      v2f b;
      b.x = Bj[(long)(ka + 0) * bs_k + (long)idx * bs_n];
      b.y = Bj[(long)(ka + 1) * bs_k + (long)idx * bs_n];
      acc[j] = wmma4(a, b, acc[j]);
    }
  }
}

// --- two independent (A,B) pairs interleaved into two accumulators ---------
__device__ __forceinline__ void wmma_tile_dual(
    const float* __restrict__ A0, const float* __restrict__ B0, long b0_k, long b0_n,
    const float* __restrict__ A1, const float* __restrict__ B1, long b1_k, long b1_n,
    long as_m, long as_k, int Ktot, v8f& y0, v8f& y1)
{
  const int lane = threadIdx.x & 31;
  const int idx  = lane & 15;
  const int half = lane >> 4;
  for (int k = 0; k < Ktot; k += 4) {
    const int ka = k + 2 * half;
    v2f a, b;
    a.x = A0[(long)idx * as_m + (long)(ka + 0) * as_k];
    a.y = A0[(long)idx * as_m + (long)(ka + 1) * as_k];
    b.x = B0[(long)(ka + 0) * b0_k + (long)idx * b0_n];
    b.y = B0[(long)(ka + 1) * b0_k + (long)idx * b0_n];
    y0 = wmma4(a, b, y0);
    a.x = A1[(long)idx * as_m + (long)(ka + 0) * as_k];
    a.y = A1[(long)idx * as_m + (long)(ka + 1) * as_k];
    b.x = B1[(long)(ka + 0) * b1_k + (long)idx * b1_n];
    b.y = B1[(long)(ka + 1) * b1_k + (long)idx * b1_n];
    y1 = wmma4(a, b, y1);
  }
}

// ---------------------------------------------------------------------------
// Generic batched WMMA GEMM, compile-time A/B strides and K.
// One wave per 16x64 C strip (4 independent accumulator chains).
// ---------------------------------------------------------------------------
template<int AS_M, int AS_K, int BS_K, int BS_N, int KTOT>
__global__ __launch_bounds__(32)
void gemm_wmma_kernel(
    const float* __restrict__ A, long a_sb, long a_sp,
    const float* __restrict__ B, long b_sb, long b_sp,
    float* __restrict__ C, long c_sb, long c_sp, long cs_m, long cs_n,
    const float* __restrict__ biasN, int relu, int Pdim)
{
  const int tn = blockIdx.x;        // 64-wide N strip
  const int tm = blockIdx.y;        // 16-wide M tile
  const int z  = blockIdx.z;
  const int b  = z / Pdim;
  const int p  = z % Pdim;
  A += (long)b * a_sb + (long)p * a_sp;
  B += (long)b * b_sb + (long)p * b_sp;
  C += (long)b * c_sb + (long)p * c_sp;

  v8f acc[4] = {v8zero(), v8zero(), v8zero(), v8zero()};
  wmma_tile4(A + (long)tm * 16 * AS_M, AS_M, AS_K,
             B + (long)tn * 64 * BS_N, BS_K, BS_N, (long)16 * BS_N,
             KTOT, acc);

  const int lane = threadIdx.x & 31;
  const int idx  = lane & 15;
  const int half = lane >> 4;
#pragma unroll
  for (int j = 0; j < 4; ++j) {
#pragma unroll
    for (int r = 0; r < 8; ++r) {
      const int m = tm * 16 + r + 8 * half;
      const int n = tn * 64 + j * 16 + idx;
      float v = acc[j][r];
      if (biasN) v += biasN[n];
      if (relu)  v = fmaxf(v, 0.f);
      C[(long)m * cs_m + (long)n * cs_n] = v;
    }
  }
}

// ---------------------------------------------------------------------------
// Conv stem: e = relu(conv1(relu(conv0(h)))) + h, per 3x16x16 image.
// ---------------------------------------------------------------------------
__global__ __launch_bounds__(256)
void conv_res_kernel(const float* __restrict__ X,
                     const float* __restrict__ w0, const float* __restrict__ b0,
                     const float* __restrict__ w1, const float* __restrict__ b1,
                     float* __restrict__ E)
{
  __shared__ float s_in[3 * 256];
  __shared__ float s_mid[32 * 256];
  __shared__ float s_w0[864];
  __shared__ float s_w1[864];
  __shared__ float s_b0[32];
  __shared__ float s_b1[3];

  const int img = blockIdx.x;
  const int t   = threadIdx.x;
  const float* xin = X + (long)img * 768;
  for (int i = t; i < 768; i += 256) s_in[i] = xin[i];
  for (int i = t; i < 864; i += 256) s_w0[i] = w0[i];
  for (int i = t; i < 864; i += 256) s_w1[i] = w1[i];
  if (t < 32) s_b0[t] = b0[t];
  if (t < 3)  s_b1[t] = b1[t];
  __syncthreads();

  const int px = t & 15, py = t >> 4;
  for (int oc = 0; oc < 32; ++oc) {
    float acc = s_b0[oc];
#pragma unroll
    for (int ic = 0; ic < 3; ++ic)
#pragma unroll
      for (int ky = 0; ky < 3; ++ky) {
        const int y = py + ky - 1;
        if (y < 0 || y > 15) continue;
#pragma unroll
        for (int kx = 0; kx < 3; ++kx) {
          const int x = px + kx - 1;
          if (x < 0 || x > 15) continue;
          acc += s_w0[((oc * 3 + ic) * 3 + ky) * 3 + kx] * s_in[ic * 256 + y * 16 + x];
        }
      }
    s_mid[oc * 256 + t] = fmaxf(acc, 0.f);
  }
  __syncthreads();

  float* eo = E + (long)img * 768;
  for (int oc = 0; oc < 3; ++oc) {
    float acc = s_b1[oc];
    for (int ic = 0; ic < 32; ++ic)
#pragma unroll
      for (int ky = 0; ky < 3; ++ky) {
        const int y = py + ky - 1;
        if (y < 0 || y > 15) continue;
#pragma unroll
        for (int kx = 0; kx < 3; ++kx) {
          const int x = px + kx - 1;
          if (x < 0 || x > 15) continue;
          acc += s_w1[((oc * 32 + ic) * 3 + ky) * 3 + kx] * s_mid[ic * 256 + y * 16 + x];
        }
      }
    eo[oc * 256 + t] = fmaxf(acc, 0.f) + s_in[oc * 256 + t];
  }
}

// ---------------------------------------------------------------------------
// Fused GAT attention (template on G, F): per block = 16 attention rows of
// one (b,p). scores -> masked softmax -> z-chunks -> wf mix, all via WMMA
// with the score panel and z-chunk staged in LDS. 8 waves / block.
// ---------------------------------------------------------------------------
template<int G, int F>
__global__ __launch_bounds__(256)
void gat_attn_kernel(
    const float* __restrict__ X,    // (B, G, N)
    const float* __restrict__ Kb,   // (B*P, F, N)
    const float* __restrict__ Qb,   // (B*P, F, N)
    const float* __restrict__ S,    // (B, 2, N, N)
    int sk,
    const float* __restrict__ WF,   // (P, 1, 2, G, F)
    const float* __restrict__ bias, // (F)
    float* __restrict__ Out,        // (B, P*F, N)
    int accum)
{
  __shared__ float s_sc[16 * SST];
  __shared__ float s_z[GCH * SZT];

  const int wave = threadIdx.x >> 5;
  const int lane = threadIdx.x & 31;
  const int idx  = lane & 15;
  const int half = lane >> 4;
  const int n0 = blockIdx.x * 16;
  const int bp = blockIdx.y;
  const int b  = bp / PP;
  const int p  = bp % PP;

  const float* Xb = X + (long)b * G * NN;
  const float* Kp = Kb + (long)bp * F * NN;
  const float* Qp = Qb + (long)bp * F * NN;
  const float* Sb = S + ((long)b * 2 + sk) * (long)NN * NN;

  // --- 1) scores panel: rows [n0,n0+16) x 512 cols; 4 m-tiles per wave ----
  {
    v8f sacc[4] = {v8zero(), v8zero(), v8zero(), v8zero()};
    // A[row][f] = Q[f][n0+row]; B tiles at m-columns (wave + 8j)*16
    wmma_tile4(Qp + n0, 1, NN, Kp + wave * 16, NN, 1, /*btile=*/8 * 16,
               F, sacc);
#pragma unroll
    for (int j = 0; j < 4; ++j) {
      const int mt = wave + 8 * j;
#pragma unroll
      for (int r = 0; r < 8; ++r) {
        const int row = r + 8 * half;
        const int col = mt * 16 + idx;
        float s = sacc[j][r];
        s = (s > 0.f) ? s : LSLOPE * s;                 // leaky_relu
        const float sv = Sb[(long)(n0 + row) * NN + col];
        if (!(fabsf(sv) > 1e-9f)) s = -1.0e9f;          // mask fill
        s_sc[row * SST + col] = s;
      }
    }
  }
  __syncthreads();

  // --- 2) softmax per row (diag always unmasked -> finite max) ------------
#pragma unroll
  for (int rr = 0; rr < 2; ++rr) {
    const int row = wave * 2 + rr;
    float mx = -1.0e30f;
    for (int m = lane; m < NN; m += 32) mx = fmaxf(mx, s_sc[row * SST + m]);
    for (int o = 16; o > 0; o >>= 1) mx = fmaxf(mx, __shfl_xor(mx, o, 32));
    float sum = 0.f;
    for (int m = lane; m < NN; m += 32) {
      const float ev = __expf(s_sc[row * SST + m] - mx);  // masked -> 0
      s_sc[row * SST + m] = ev;
      sum += ev;
    }
    for (int o = 16; o > 0; o >>= 1) sum += __shfl_xor(sum, o, 32);
    const float inv = 1.f / sum;
    for (int m = lane; m < NN; m += 32) s_sc[row * SST + m] *= inv;
  }
  __syncthreads();

  // --- 3) propagate + feature mix over 128-wide g chunks ------------------
  v8f y0 = v8zero(), y1 = v8zero();
  const int nyt = F >> 4;                 // y tiles, one per wave (<=8)
  for (int g0 = 0; g0 < G; g0 += GCH) {
    {
      // z tile (16 g x 16 n): A[g][m]=X[g0+wave*16+g][m]; B[m][n]=aij[n][m]
      v8f zacc = wmma_tile_split2(Xb + (long)(g0 + wave * 16) * NN, NN, 1,
                                  s_sc, 1, SST, NN, v8zero());
#pragma unroll
      for (int r = 0; r < 8; ++r)
        s_z[(wave * 16 + r + 8 * half) * SZT + idx] = zacc[r];
    }
    __syncthreads();
    if (wave < nyt) {
      // y0 += Wf0^T X[:,ntile];  y1 += Wf1^T z   (independent chains)
      const float* A0 = WF + ((long)(p * 2 + 0) * G + g0) * F + wave * 16;
      const float* A1 = WF + ((long)(p * 2 + 1) * G + g0) * F + wave * 16;
      wmma_tile_dual(A0, Xb + (long)g0 * NN + n0, NN, 1,
                     A1, s_z, SZT, 1,
                     1, F, GCH, y0, y1);
    }
    __syncthreads();
  }

  // --- 4) epilogue --------------------------------------------------------
  if (wave < nyt) {
#pragma unroll
    for (int r = 0; r < 8; ++r) {
      const int f = wave * 16 + r + 8 * half;
      const int n = n0 + idx;
      float v = fmaxf(y0[r] + y1[r] + bias[f], 0.f);
      const long o = ((long)b * (PP * F) + (long)p * F + f) * NN + n;
      if (accum) Out[o] += v; else Out[o] = v;
    }
  }
}

// ---------------------------------------------------------------------------
// Decoder tail: out[r,:] = softmax(dhid[r,:] @ dec1_w^T + dec1_b), 5 classes.
// ---------------------------------------------------------------------------
__global__ __launch_bounds__(256)
void dec1_softmax_kernel(const float* __restrict__ Hd,
                         const float* __restrict__ W,
                         const float* __restrict__ bb,
                         float* __restrict__ Out)
{
  const int r = blockIdx.x * blockDim.x + threadIdx.x;
  if (r >= 4096) return;
  const float* h = Hd + (long)r * 256;
  float v[5];
  float mx = -1.0e30f;
#pragma unroll
  for (int j = 0; j < 5; ++j) {
    float a = bb[j];
    const float* w = W + j * 256;
    for (int k = 0; k < 256; ++k) a += h[k] * w[k];
    v[j] = a;
    mx = fmaxf(mx, a);
  }
  float s = 0.f;
#pragma unroll
  for (int j = 0; j < 5; ++j) { v[j] = __expf(v[j] - mx); s += v[j]; }
  const float inv = 1.f / s;
#pragma unroll
  for (int j = 0; j < 5; ++j) Out[(long)r * 5 + j] = v[j] * inv;
}

// ---------------------------------------------------------------------------
template<int G, int F>
static inline void run_gat_layer(const float* xin,
                                 const float* wk, const float* wq,
                                 const float* wf, const float* bb,
                                 const float* S, int sk,
                                 float* Kb, float* Qb,
                                 float* outp, int accum, hipStream_t stream)
{
  dim3 gk(NN / 64, F / 16, 8 * PP);
  // K[bp] = Wk^T(FxG) * X[b](GxN);  Q[bp] = Wq^T * X[b]
  gemm_wmma_kernel<1, F, NN, 1, G><<<gk, 32, 0, stream>>>(
      wk, 0, (long)G * F, xin, (long)G * NN, 0,
      Kb, (long)PP * F * NN, (long)F * NN, NN, 1, nullptr, 0, PP);
  gemm_wmma_kernel<1, F, NN, 1, G><<<gk, 32, 0, stream>>>(
      wq, 0, (long)G * F, xin, (long)G * NN, 0,
      Qb, (long)PP * F * NN, (long)F * NN, NN, 1, nullptr, 0, PP);
  gat_attn_kernel<G, F><<<dim3(NN / 16, 8 * PP), 256, 0, stream>>>(
      xin, Kb, Qb, S, sk, wf, bb, outp, accum);
}

extern "C" void kernel_launch(void* const* d_in, const int* in_sizes, int n_in,
                              void* d_out, int out_size, void* d_ws, size_t ws_size,
                              hipStream_t stream) {
  (void)in_sizes; (void)n_in; (void)out_size; (void)ws_size;
  const float* x     = (const float*)d_in[0];
  const float* S     = (const float*)d_in[1];
  const float* c0w   = (const float*)d_in[2];
  const float* c0b   = (const float*)d_in[3];
  const float* c1w   = (const float*)d_in[4];
  const float* c1b   = (const float*)d_in[5];
  const float* fw    = (const float*)d_in[6];
  const float* fb    = (const float*)d_in[7];
  const float* dec0w = (const float*)d_in[32];
  const float* dec0b = (const float*)d_in[33];
  const float* dec1w = (const float*)d_in[34];
  const float* dec1b = (const float*)d_in[35];
  float* out = (float*)d_out;

  // Workspace (floats). All reference reshapes are reinterprets of contiguous
  // row-major memory, so f==g and u1==decoder-input with no shuffles.
  float* ws = (float*)d_ws;
  float* p0 = ws;                           // (4096,128)  == g (8,128,512)
  float* p1 = p0 + 4096l * 128;             // (8,512,512)
  float* p2 = p1 + 8l * 512 * 512;          // (8,128,512)
  float* u0 = p2 + 4096l * 128;             // (8,256,512)
  float* u1 = u0 + 8l * 256 * 512;          // (8,512,512) == (4096,512)
  float* Kb = u1 + 8l * 512 * 512;          // (32,Fmax=128,512)
  float* Qb = Kb + 32l * 128 * 512;         // (32,128,512)
  float* e  = Kb;                           // (4096,768) alias: dead before GATs
  float* dh = Kb;                           // (4096,256) alias: used after GATs

  // 1) conv stem + residual
  conv_res_kernel<<<4096, 256, 0, stream>>>(x, c0w, c0b, c1w, c1b, e);

  // 2) p0 = relu(e(4096x768) @ fw^T(768x128) + fb)   [unit-k strides -> b64]
  gemm_wmma_kernel<768, 1, 1, 768, 768><<<dim3(128 / 64, 4096 / 16, 1), 32, 0, stream>>>(
      e, 0, 0, fw, 0, 0, p0, 0, 0, 128, 1, fb, 1, 1);

  // 3) GAT stack
  run_gat_layer<128, 128>(p0, (const float*)d_in[ 8], (const float*)d_in[ 9],
                          (const float*)d_in[10], (const float*)d_in[11],
                          S, 0, Kb, Qb, p1, 0, stream);            // d0
  run_gat_layer<512,  32>(p1, (const float*)d_in[12], (const float*)d_in[13],
                          (const float*)d_in[14], (const float*)d_in[15],
                          S, 1, Kb, Qb, p2, 0, stream);            // d1
  run_gat_layer<128,  64>(p2, (const float*)d_in[16], (const float*)d_in[17],
                          (const float*)d_in[18], (const float*)d_in[19],
                          S, 1, Kb, Qb, u0, 0, stream);            // u0
  run_gat_layer<512,  64>(p1, (const float*)d_in[28], (const float*)d_in[29],
                          (const float*)d_in[30], (const float*)d_in[31],
                          S, 1, Kb, Qb, u0, 1, stream);            // += s1
  run_gat_layer<256, 128>(u0, (const float*)d_in[20], (const float*)d_in[21],
                          (const float*)d_in[22], (const float*)d_in[23],
                          S, 0, Kb, Qb, u1, 0, stream);            // u1
  run_gat_layer<128, 128>(p0, (const float*)d_in[24], (const float*)d_in[25],
                          (const float*)d_in[26], (const float*)d_in[27],
                          S, 0, Kb, Qb, u1, 1, stream);            // += s0

  // 4) dh = relu(u1(4096x512) @ dec0w^T(512x256) + dec0b)  [unit-k -> b64]
  gemm_wmma_kernel<512, 1, 1, 512, 512><<<dim3(256 / 64, 4096 / 16, 1), 32, 0, stream>>>(
      u1, 0, 0, dec0w, 0, 0, dh, 0, 0, 256, 1, dec0b, 1, 1);

  // 5) out = softmax(dh @ dec1w^T + dec1b)
  dec1_softmax_kernel<<<16, 256, 0, stream>>>(dh, dec1w, dec1b, out);
}